// HybridBlockRecurrentMamba_40046275068366
// MI455X (gfx1250) — compile-verified
//
#include <hip/hip_runtime.h>
#include <hip/hip_bf16.h>

// ---------------- model constants ----------------
#define BB   2
#define LL   2048
#define DM   512
#define DI   1024
#define DS   16
#define DTR  32
#define NL   15
#define BLK  64
#define NH   4
#define CK   4
#define NB   (LL / BLK)    // 32 chunks
#define TT   (BLK + 1)     // 65 rows per chunk (state token + 64)
#define MR   (BB * TT)     // 130 GEMM rows per layer-chunk

typedef __attribute__((ext_vector_type(2))) float v2f;
typedef __attribute__((ext_vector_type(8))) float v8f;

enum { EPI_NONE = 0, EPI_SOFTPLUS = 1 };

// =====================================================================
// fp32 WMMA GEMM:  C[M,N] = A[M,K(lda)] * B  (+bias) (+epilogue)
// B is Bm[K(ldb),N] if TRANSB==0, else Bm[N(ldb),K].
// Requirements: N % 64 == 0, K % 16 == 0. M arbitrary (clamped rows;
// only rows < M are stored, so out-of-range lanes read row M-1).
// Block = 128 threads (4 waves); each wave owns one 16x16 tile,
// block covers 16(M) x 64(N); grid = (N/64, ceil(M/16)).
// A subtile (16xK) staged in LDS per 16-K step, shared by all 4 waves.
// For TRANSB, B subtile (64x16) is also LDS-staged (lane-strided otherwise).
// =====================================================================
template <int TRANSB, int EPI>
__global__ __launch_bounds__(128)
void gemm_f32_wmma(const float* __restrict__ A, const float* __restrict__ Bm,
                   const float* __restrict__ bias, float* __restrict__ C,
                   int M, int N, int K, int lda, int ldb, int ldc)
{
    const int tid   = threadIdx.x;
    const int lane  = tid & 31;
    const int wave  = tid >> 5;
    const int r     = lane & 15;
    const int half  = lane >> 4;            // 0: lanes 0-15, 1: lanes 16-31
    const int nBase = blockIdx.x * 64 + wave * 16;
    const int mBase = blockIdx.y * 16;

    __shared__ float sA[16][18];            // pad 18: conflict-free, 8B-aligned pairs
    __shared__ float sB[TRANSB ? 64 : 1][18];

    // A staging addresses: thread loads 2 consecutive floats of one row
    const int aRow = tid >> 3;              // (tid*2)>>4 : 0..15
    const int aCol = (tid & 7) * 2;         // 0,2,..,14
    const int mClamp = min(mBase + aRow, M - 1);
    const float* Ars = A + (size_t)mClamp * lda + aCol;

    // B staging (TRANSB): thread loads 8 consecutive floats of one row
    const int bRow = tid >> 1;              // 0..63
    const int bCol = (tid & 1) * 8;         // 0 or 8
    const float* Brs = TRANSB ? (Bm + (size_t)(blockIdx.x * 64 + bRow) * ldb + bCol)
                              : nullptr;

    const float* Bp0 = TRANSB ? nullptr : (Bm + nBase + r);

    v8f acc = {};
    for (int k0 = 0; k0 < K; k0 += 16) {
        __syncthreads();
        // ---- stage A 16x16 subtile (coalesced) ----
        {
            v2f t = *(const v2f*)(Ars + k0);
            sA[aRow][aCol]     = t.x;
            sA[aRow][aCol + 1] = t.y;
        }
        // ---- stage B 64x16 subtile when TRANSB ----
        if (TRANSB) {
#pragma unroll
            for (int i = 0; i < 8; ++i)
                sB[bRow][bCol + i] = Brs[k0 + i];
        }
        __syncthreads();

        // ---- 4 WMMA steps of K=4 each ----
        v2f a[4], b[4];
#pragma unroll
        for (int u = 0; u < 4; ++u) {
            const int ka = 4 * u + 2 * half;    // lanes 0-15: K{+0,+1}; 16-31: K{+2,+3}
            a[u] = *(const v2f*)(&sA[r][ka]);
            if (TRANSB) {
                b[u] = *(const v2f*)(&sB[wave * 16 + r][ka]);
            } else {
                const float* Bp = Bp0 + (size_t)(k0 + ka) * ldb;
                b[u].x = Bp[0];
                b[u].y = Bp[ldb];
            }
        }
#pragma unroll
        for (int u = 0; u < 4; ++u)
            acc = __builtin_amdgcn_wmma_f32_16x16x4_f32(
                      false, a[u], false, b[u], (short)0, acc, false, false);
    }

    const int   n  = nBase + r;
    const float bv = bias ? bias[n] : 0.0f;
#pragma unroll
    for (int i = 0; i < 8; ++i) {
        const int mi = mBase + i + 8 * half;    // C/D: VGPR i -> row i (+8 for hi half)
        if (mi < M) {
            float v = acc[i] + bv;
            if (EPI == EPI_SOFTPLUS) v = (v > 20.0f) ? v : log1pf(expf(v));
            C[(size_t)mi * ldc + n] = v;
        }
    }
}

// ---------------- embedding gather ----------------
__global__ void embed_kernel(const int* __restrict__ tok, const float* __restrict__ emb,
                             float* __restrict__ h)
{
    int idx = blockIdx.x * blockDim.x + threadIdx.x;
    if (idx >= BB * LL * DM) return;
    int i = idx & (DM - 1);
    int t = idx >> 9;                 // b*L + l
    h[idx] = emb[(size_t)tok[t] * DM + i];
}

// ---------------- init per-layer states from init_state (broadcast over B) -------
__global__ void init_states_kernel(const float* __restrict__ init_state, float* __restrict__ st)
{
    int idx = blockIdx.x * blockDim.x + threadIdx.x;
    if (idx >= NL * BB * DM) return;
    int i = idx & (DM - 1);
    int j = idx / (BB * DM);
    st[idx] = init_state[(size_t)j * DM + i];
}

// ---------------- rmsnorm + concat(state, rmsnorm(co chunk)) -> cwc (B,65,512) ----
__global__ __launch_bounds__(128)
void rmsnorm_concat_kernel(const float* __restrict__ st_j, const float* __restrict__ co,
                           const float* __restrict__ rw, float* __restrict__ cwc, int chunk)
{
    const int row = blockIdx.x;           // b*65 + t
    const int b = row / TT, t = row % TT;
    const int tid = threadIdx.x;
    float* dst = cwc + (size_t)row * DM;
    if (t == 0) {
        for (int i = tid; i < DM; i += 128) dst[i] = st_j[(size_t)b * DM + i];
        return;
    }
    const float* src = co + ((size_t)b * LL + chunk * BLK + (t - 1)) * DM;
    __shared__ float red[128];
    float ss = 0.0f;
    for (int i = tid; i < DM; i += 128) { float v = src[i]; ss += v * v; }
    red[tid] = ss; __syncthreads();
    for (int off = 64; off > 0; off >>= 1) {
        if (tid < off) red[tid] += red[tid + off];
        __syncthreads();
    }
    const float inv = rsqrtf(red[0] / (float)DM + 1e-6f);
    for (int i = tid; i < DM; i += 128) dst[i] = src[i] * inv * rw[i];
}

// ---------------- depthwise conv (K=4, left zero pad) + SiLU ----------------
__global__ void conv_silu_kernel(const float* __restrict__ xz, const float* __restrict__ cw,
                                 const float* __restrict__ cb, float* __restrict__ xc)
{
    int idx = blockIdx.x * blockDim.x + threadIdx.x;   // B*65*1024
    if (idx >= BB * TT * DI) return;
    const int d = idx & (DI - 1);
    const int t = (idx >> 10) % TT;
    const int b = idx / (TT * DI);
    float acc = cb[d];
#pragma unroll
    for (int k = 0; k < CK; ++k) {
        const int ts = t - (CK - 1) + k;
        if (ts >= 0) acc += xz[((size_t)(b * TT + ts)) * (2 * DI) + d] * cw[d * CK + k];
    }
    xc[idx] = acc / (1.0f + expf(-acc));               // silu
}

// ---------------- selective scan + D skip + SiLU(z) gate -> ygated ----------------
__global__ __launch_bounds__(256)
void scan_kernel(const float* __restrict__ dt, const float* __restrict__ xdbl,
                 const float* __restrict__ xc, const float* __restrict__ xz,
                 const float* __restrict__ Alog, const float* __restrict__ Dsk,
                 float* __restrict__ yg)
{
    const int ch = blockIdx.x * 256 + threadIdx.x;     // B*DI channels; b uniform per block
    const int b = ch >> 10, d = ch & (DI - 1);
    __shared__ float sB[DS], sC[DS];
    float Ar[DS];
#pragma unroll
    for (int n = 0; n < DS; ++n) Ar[n] = -expf(Alog[(size_t)d * DS + n]);
    const float Dv = Dsk[d];
    float s[DS];
#pragma unroll
    for (int n = 0; n < DS; ++n) s[n] = 0.0f;

    for (int t = 0; t < TT; ++t) {
        const size_t row = (size_t)(b * TT + t);
        if (threadIdx.x < 16)      sB[threadIdx.x]      = xdbl[row * 64 + DTR      + threadIdx.x];
        else if (threadIdx.x < 32) sC[threadIdx.x - 16] = xdbl[row * 64 + DTR + DS + (threadIdx.x - 16)];
        __syncthreads();
        const float dtv = dt[row * DI + d];
        const float xv  = xc[row * DI + d];
        float y = 0.0f;
#pragma unroll
        for (int n = 0; n < DS; ++n) {
            s[n] = s[n] * expf(dtv * Ar[n]) + dtv * xv * sB[n];
            y += s[n] * sC[n];
        }
        y += Dv * xv;
        const float zv = xz[row * (2 * DI) + DI + d];
        yg[row * DI + d] = y * (zv / (1.0f + expf(-zv)));
        __syncthreads();
    }
}

// ---------------- residual add, state update, memory bank ----------------
__global__ void layer_epilogue_kernel(const float* __restrict__ lo, float* __restrict__ co,
                                      float* __restrict__ st_j, float* __restrict__ membank,
                                      int chunk, int isLast)
{
    int idx = blockIdx.x * blockDim.x + threadIdx.x;   // B*65*512
    if (idx >= BB * TT * DM) return;
    const int i = idx & (DM - 1);
    const int t = (idx >> 9) % TT;
    const int b = idx / (TT * DM);
    const float v = lo[idx];
    if (t >= 1) co[((size_t)b * LL + chunk * BLK + (t - 1)) * DM + i] += v;
    if (t == BLK) {
        st_j[(size_t)b * DM + i] = v;
        if (isLast) membank[((size_t)b * NB + chunk) * DM + i] = v;
    }
}

// ---------------- block attention over memory bank (wave32 softmax) ----------------
__global__ __launch_bounds__(128)
void attention_kernel(const float* __restrict__ q, const float* __restrict__ kb,
                      const float* __restrict__ vb, float* __restrict__ ao)
{
    const int bl = blockIdx.x;                 // b*L + l
    const int b = bl >> 11, l = bl & (LL - 1);
    const int tid = threadIdx.x;
    const int h = tid >> 5, m = tid & 31;      // one wave per head, lane = memory slot
    const int dh = DM / NH;                    // 128

    const float* qr = q  + (size_t)bl * DM + h * dh;
    const float* kr = kb + ((size_t)b * NB + m) * DM + h * dh;
    float sc = 0.0f;
    for (int i = 0; i < dh; ++i) sc += qr[i] * kr[i];
    sc *= 0.08838834764831845f;                // 1/sqrt(128)
    const int tb = l >> 6;
    sc += ((m < tb) || (tb == 0)) ? 0.0f : -1.0e9f;

    float mx = sc;
    for (int off = 16; off > 0; off >>= 1) mx = fmaxf(mx, __shfl_xor(mx, off, 32));
    const float e = expf(sc - mx);
    float sum = e;
    for (int off = 16; off > 0; off >>= 1) sum += __shfl_xor(sum, off, 32);

    __shared__ float sw[128];
    sw[tid] = e / sum;
    __syncthreads();

#pragma unroll
    for (int i = 0; i < 4; ++i) {
        const int g  = tid + i * 128;          // output column 0..511
        const int hh = g >> 7;
        float acc = 0.0f;
        for (int mm = 0; mm < NB; ++mm)
            acc += sw[hh * 32 + mm] * vb[((size_t)b * NB + mm) * DM + g];
        ao[(size_t)bl * DM + g] = acc;
    }
}

// ---------------- hres = mamba_out + ao_proj * has_past ----------------
__global__ void hres_fuse_kernel(const float* __restrict__ mo, const float* __restrict__ aop,
                                 float* __restrict__ hres)
{
    int idx = blockIdx.x * blockDim.x + threadIdx.x;
    if (idx >= BB * LL * DM) return;
    const int l = (idx >> 9) & (LL - 1);
    const float hp = (l >> 6) ? 1.0f : 0.0f;
    hres[idx] = mo[idx] + aop[idx] * hp;
}

// ---------------- LayerNorm ----------------
__global__ __launch_bounds__(128)
void layernorm_kernel(const float* __restrict__ x, const float* __restrict__ w,
                      const float* __restrict__ bsh, float* __restrict__ y)
{
    const int row = blockIdx.x;
    const float* xr = x + (size_t)row * DM;
    __shared__ float red[128];
    const int tid = threadIdx.x;
    float s = 0.0f;
    for (int i = tid; i < DM; i += 128) s += xr[i];
    red[tid] = s; __syncthreads();
    for (int off = 64; off > 0; off >>= 1) { if (tid < off) red[tid] += red[tid + off]; __syncthreads(); }
    const float mu = red[0] / (float)DM;
    __syncthreads();
    float v = 0.0f;
    for (int i = tid; i < DM; i += 128) { float d = xr[i] - mu; v += d * d; }
    red[tid] = v; __syncthreads();
    for (int off = 64; off > 0; off >>= 1) { if (tid < off) red[tid] += red[tid + off]; __syncthreads(); }
    const float inv = rsqrtf(red[0] / (float)DM + 1e-5f);
    for (int i = tid; i < DM; i += 128)
        y[(size_t)row * DM + i] = (xr[i] - mu) * inv * w[i] + bsh[i];
}

// =====================================================================
static inline void launch_gemm(const float* A, const float* Bm, const float* bias, float* C,
                               int M, int N, int K, int lda, int ldb, int ldc,
                               int transB, int epi, hipStream_t s)
{
    dim3 grid(N / 64, (M + 15) / 16);
    if (transB) {
        if (epi == EPI_SOFTPLUS)
            gemm_f32_wmma<1, EPI_SOFTPLUS><<<grid, 128, 0, s>>>(A, Bm, bias, C, M, N, K, lda, ldb, ldc);
        else
            gemm_f32_wmma<1, EPI_NONE><<<grid, 128, 0, s>>>(A, Bm, bias, C, M, N, K, lda, ldb, ldc);
    } else {
        if (epi == EPI_SOFTPLUS)
            gemm_f32_wmma<0, EPI_SOFTPLUS><<<grid, 128, 0, s>>>(A, Bm, bias, C, M, N, K, lda, ldb, ldc);
        else
            gemm_f32_wmma<0, EPI_NONE><<<grid, 128, 0, s>>>(A, Bm, bias, C, M, N, K, lda, ldb, ldc);
    }
}

extern "C" void kernel_launch(void* const* d_in, const int* in_sizes, int n_in,
                              void* d_out, int out_size, void* d_ws, size_t ws_size,
                              hipStream_t stream)
{
    const int*   tokens     = (const int*)  d_in[0];
    const float* embed      = (const float*)d_in[1];
    const float* rms_w      = (const float*)d_in[2];
    const float* W_in       = (const float*)d_in[3];
    const float* conv_w     = (const float*)d_in[4];
    const float* conv_b     = (const float*)d_in[5];
    const float* x_proj     = (const float*)d_in[6];
    const float* dt_w       = (const float*)d_in[7];
    const float* dt_b       = (const float*)d_in[8];
    const float* A_log      = (const float*)d_in[9];
    const float* D_skip     = (const float*)d_in[10];
    const float* W_out      = (const float*)d_in[11];
    const float* init_state = (const float*)d_in[12];
    const float* Wq         = (const float*)d_in[13];
    const float* Wk         = (const float*)d_in[14];
    const float* Wv         = (const float*)d_in[15];
    const float* bq         = (const float*)d_in[16];
    const float* bk         = (const float*)d_in[17];
    const float* bv         = (const float*)d_in[18];
    const float* Wo_attn    = (const float*)d_in[19];
    const float* bo         = (const float*)d_in[20];
    const float* ln_w       = (const float*)d_in[21];
    const float* ln_b       = (const float*)d_in[22];
    const float* head_w     = (const float*)d_in[23];
    float* out = (float*)d_out;

    // -------- workspace carve-up (floats) --------
    float* p = (float*)d_ws;
    float* states = p; p += (size_t)NL * BB * DM;
    float* mamba  = p; p += (size_t)BB * LL * DM;     // mamba_out / residual stream
    float* cwc    = p; p += (size_t)BB * TT * DM;
    float* xz     = p; p += (size_t)BB * TT * 2 * DI;
    float* xc     = p; p += (size_t)BB * TT * DI;
    float* xdbl   = p; p += (size_t)BB * TT * 64;
    float* dtb    = p; p += (size_t)BB * TT * DI;
    float* yg     = p; p += (size_t)BB * TT * DI;
    float* lo     = p; p += (size_t)BB * TT * DM;
    float* mem    = p; p += (size_t)BB * NB * DM;
    float* qb     = p; p += (size_t)BB * LL * DM;     // q, later reused for hres
    float* kb     = p; p += (size_t)BB * NB * DM;
    float* vb     = p; p += (size_t)BB * NB * DM;
    float* ao     = p; p += (size_t)BB * LL * DM;     // attn out, later reused for hf
    float* aop    = p; p += (size_t)BB * LL * DM;     // attn out projected
    (void)ws_size; (void)in_sizes; (void)n_in; (void)out_size;

    // -------- embedding + state init --------
    embed_kernel<<<(BB * LL * DM + 255) / 256, 256, 0, stream>>>(tokens, embed, mamba);
    init_states_kernel<<<(NL * BB * DM + 255) / 256, 256, 0, stream>>>(init_state, states);

    // -------- sequential chunk x layer recurrence --------
    for (int ci = 0; ci < NB; ++ci) {
        for (int j = 0; j < NL; ++j) {
            const float* stj = states + (size_t)j * BB * DM;
            rmsnorm_concat_kernel<<<BB * TT, 128, 0, stream>>>(stj, mamba, rms_w + (size_t)j * DM, cwc, ci);
            // xz = cwc @ W_in[j]            (130 x 2048, K=512)
            launch_gemm(cwc, W_in + (size_t)j * DM * 2 * DI, nullptr, xz,
                        MR, 2 * DI, DM, DM, 2 * DI, 2 * DI, 0, EPI_NONE, stream);
            conv_silu_kernel<<<(BB * TT * DI + 255) / 256, 256, 0, stream>>>(
                xz, conv_w + (size_t)j * DI * CK, conv_b + (size_t)j * DI, xc);
            // xdbl = xc @ x_proj[j]         (130 x 64, K=1024)
            launch_gemm(xc, x_proj + (size_t)j * DI * 64, nullptr, xdbl,
                        MR, 64, DI, DI, 64, 64, 0, EPI_NONE, stream);
            // dt = softplus(xdbl[:, :32] @ dt_w[j] + dt_b[j])   (130 x 1024, K=32)
            launch_gemm(xdbl, dt_w + (size_t)j * DTR * DI, dt_b + (size_t)j * DI, dtb,
                        MR, DI, DTR, 64, DI, DI, 0, EPI_SOFTPLUS, stream);
            scan_kernel<<<(BB * DI) / 256, 256, 0, stream>>>(
                dtb, xdbl, xc, xz, A_log + (size_t)j * DI * DS, D_skip + (size_t)j * DI, yg);
            // lo = yg @ W_out[j]            (130 x 512, K=1024)
            launch_gemm(yg, W_out + (size_t)j * DI * DM, nullptr, lo,
                        MR, DM, DI, DI, DM, DM, 0, EPI_NONE, stream);
            layer_epilogue_kernel<<<(BB * TT * DM + 255) / 256, 256, 0, stream>>>(
                lo, mamba, states + (size_t)j * BB * DM, mem, ci, (j == NL - 1) ? 1 : 0);
        }
    }

    // -------- attention over memory bank --------
    launch_gemm(mamba, Wq, bq, qb, BB * LL, DM, DM, DM, DM, DM, 1, EPI_NONE, stream);
    launch_gemm(mem,   Wk, bk, kb, BB * NB, DM, DM, DM, DM, DM, 1, EPI_NONE, stream);
    launch_gemm(mem,   Wv, bv, vb, BB * NB, DM, DM, DM, DM, DM, 1, EPI_NONE, stream);
    attention_kernel<<<BB * LL, 128, 0, stream>>>(qb, kb, vb, ao);
    launch_gemm(ao, Wo_attn, bo, aop, BB * LL, DM, DM, DM, DM, DM, 1, EPI_NONE, stream);

    // -------- residual + LN + head --------
    float* hres = qb;    // q no longer needed
    hres_fuse_kernel<<<(BB * LL * DM + 255) / 256, 256, 0, stream>>>(mamba, aop, hres);
    float* hf = ao;      // raw attn out no longer needed
    layernorm_kernel<<<BB * LL, 128, 0, stream>>>(hres, ln_w, ln_b, hf);
    launch_gemm(hf, head_w, nullptr, out, BB * LL, DM, DM, DM, DM, DM, 0, EPI_NONE, stream);
}